// AdditiveScorer_12532714569859
// MI455X (gfx1250) — compile-verified
//
#include <hip/hip_runtime.h>

typedef __attribute__((ext_vector_type(2))) float v2f;
typedef __attribute__((ext_vector_type(8))) float v8f;

#define BSZ  8
#define LQ   512
#define LK   512
#define DIN  512
#define HDIM 128

// ---------------- hardware tanh (CDNA5 V_TANH_F32) ----------------
__device__ __forceinline__ float htanh(float x) {
#if __has_builtin(__builtin_amdgcn_tanhf)
  return __builtin_amdgcn_tanhf(x);
#else
  float y;
  // TRANS op: 1 V_NOP after trans before result use (ISA §7.4 case 3)
  asm volatile("v_tanh_f32 %0, %1\n\tv_nop" : "=v"(y) : "v"(x));
  return y;
#endif
}

// ---------------- projection GEMM: C[M,HDIM] = A[M,DIN] * W[DIN,HDIM] ----------------
// One wave computes a 16x16 tile of C via V_WMMA_F32_16X16X4_F32, looping K in steps of 4.
// A frag (16x4 f32): lanes 0-15 -> row=lane, VGPR{0,1}=K{0,1}; lanes 16-31 -> K{2,3}.
// B frag (4x16 f32): lanes 0-15 -> col=lane, VGPR{0,1}=K{0,1}; lanes 16-31 -> K{2,3}.
// C/D (16x16 f32): VGPR i, lanes 0-15 -> M=i, N=lane; lanes 16-31 -> M=i+8, N=lane-16.
__global__ __launch_bounds__(128) void proj_gemm(const float* __restrict__ A,
                                                 const float* __restrict__ W,
                                                 float* __restrict__ C) {
  const int wave = threadIdx.x >> 5;
  const int lane = threadIdx.x & 31;
  const int half = lane >> 4;   // 0 or 1
  const int lr   = lane & 15;

  const int tilesN = HDIM / 16;              // 8
  const int tile   = blockIdx.x * 4 + wave;  // 4 waves/block
  const int tm     = (tile / tilesN) * 16;   // output row base
  const int tn     = (tile % tilesN) * 16;   // output col base

  const float* arow = A + (size_t)(tm + lr) * DIN + half * 2;  // 8B aligned
  const float* bcol = W + (size_t)(half * 2) * HDIM + (tn + lr);

  v8f c = {};
  for (int k = 0; k < DIN; k += 4) {
    v2f a = *(const v2f*)(arow + k);         // A[m][k0+2h], A[m][k0+2h+1]
    v2f b;
    b.x = bcol[(size_t)k * HDIM];            // W[k0+2h  ][n]
    b.y = bcol[(size_t)k * HDIM + HDIM];     // W[k0+2h+1][n]
    // 8 args: (neg_a, A, neg_b, B, c_mod, C, reuse_a, reuse_b)
    c = __builtin_amdgcn_wmma_f32_16x16x4_f32(false, a, false, b, (short)0, c,
                                              false, false);
  }

  float* crow = C + (size_t)(tm + half * 8) * HDIM + tn + lr;
#pragma unroll
  for (int i = 0; i < 8; ++i) crow[(size_t)i * HDIM] = c[i];
}

// ---------------- scoring: out[b,q,k] = sum_h wv[h] * tanh(Q[b,q,h] + K[b,k,h]) ----------------
// Block = 256 threads (8 waves). Wave w: q-row q0+w. Lane j: k-col k0+j.
// K tile padded to 129 words/row => column reads hit banks (j+h)%64, conflict-free.
__global__ __launch_bounds__(256) void score_kernel(const float* __restrict__ Qp,
                                                    const float* __restrict__ Kp,
                                                    const float* __restrict__ wv,
                                                    float* __restrict__ out) {
  __shared__ float Kt[32][129];
  __shared__ float Qt[8][128];
  __shared__ float wvs[128];

  const int b  = blockIdx.z;
  const int q0 = blockIdx.y * 8;
  const int k0 = blockIdx.x * 32;
  const int t  = threadIdx.x;

  for (int idx = t; idx < 32 * 128; idx += 256) {
    int r = idx >> 7, cc = idx & 127;
    Kt[r][cc] = Kp[(size_t)(b * LK + k0 + r) * HDIM + cc];
  }
  for (int idx = t; idx < 8 * 128; idx += 256) {
    int r = idx >> 7, cc = idx & 127;
    Qt[r][cc] = Qp[(size_t)(b * LQ + q0 + r) * HDIM + cc];
  }
  if (t < 128) wvs[t] = wv[t];
  __syncthreads();

  const int w = t >> 5;  // q row within tile
  const int j = t & 31;  // k col within tile

  float acc = 0.f;
#pragma unroll 8
  for (int h = 0; h < HDIM; ++h) {
    acc += wvs[h] * htanh(Qt[w][h] + Kt[j][h]);
  }

  out[(size_t)(b * LQ + q0 + w) * LK + k0 + j] = acc;
}

// ---------------- launch ----------------
extern "C" void kernel_launch(void* const* d_in, const int* in_sizes, int n_in,
                              void* d_out, int out_size, void* d_ws, size_t ws_size,
                              hipStream_t stream) {
  (void)in_sizes; (void)n_in; (void)out_size; (void)ws_size;
  const float* qs = (const float*)d_in[0];  // [B,LQ,D]
  const float* ks = (const float*)d_in[1];  // [B,LK,D]
  const float* Wq = (const float*)d_in[2];  // [D,H]
  const float* Wk = (const float*)d_in[3];  // [D,H]
  const float* wv = (const float*)d_in[4];  // [H]
  float* out = (float*)d_out;               // [B,LQ,LK]

  float* Qp = (float*)d_ws;                       // [B*LQ, H] = 2 MB
  float* Kp = Qp + (size_t)BSZ * LQ * HDIM;       // [B*LK, H] = 2 MB

  const int M     = BSZ * LQ;                     // 4096
  const int tiles = (M / 16) * (HDIM / 16);       // 2048 tiles, 4 waves/block
  dim3 gblk(tiles / 4);

  proj_gemm<<<gblk, 128, 0, stream>>>(qs, Wq, Qp);
  proj_gemm<<<gblk, 128, 0, stream>>>(ks, Wk, Kp);

  dim3 sgrid(LK / 32, LQ / 8, BSZ);               // 16 x 64 x 8
  score_kernel<<<sgrid, 256, 0, stream>>>(Qp, Kp, wv, out);
}